// NodePropagatorSparse_48430051230129
// MI455X (gfx1250) — compile-verified
//
#include <hip/hip_runtime.h>
#include <stdint.h>

// ---------------- problem constants (from reference) ----------------
constexpr int B_ = 4;
constexpr int N_ = 10000;
constexpr int E_ = 160000;
constexpr int D_ = 256;                  // floats per node row
// row in  = D*4      = 1024 bytes ; row out = 2*D*4 = 2048 bytes

#define WAVE_SZ          32
#define WAVES_PER_BLOCK  8
#define BLOCK_THREADS    (WAVES_PER_BLOCK * WAVE_SZ)
#define GRID_BLOCKS      1024

// ---------------- CDNA5 async global<->LDS helpers ----------------
// Builtin signature (confirmed by round-1 diagnostic):
//   void __builtin_amdgcn_global_load_async_to_lds_b128(
//        v4i addrspace(1)* gsrc, v4i addrspace(3)* ldsdst, imm offset, imm cpol)
typedef int v4i __attribute__((vector_size(16)));
typedef __attribute__((address_space(1))) v4i glb_v4i;
typedef __attribute__((address_space(3))) v4i lds_v4i;

__device__ __forceinline__ glb_v4i* to_glb(const void* p) {
  return (glb_v4i*)(uintptr_t)p;
}
__device__ __forceinline__ lds_v4i* to_lds(void* p) {
  // generic LDS address: low 32 bits are the LDS byte offset
  return (lds_v4i*)(uint32_t)(uintptr_t)p;
}

// cpol: TH in bits [2:0].  Stores use TH=NT(1) so the 1.31 GB output stream
// does not evict the 41 MB L2-resident node table (keeps gather reads in L2).
#define CPOL_RT 0
#define CPOL_NT 1

#if defined(__AMDGCN__) && \
    __has_builtin(__builtin_amdgcn_global_load_async_to_lds_b128) && \
    __has_builtin(__builtin_amdgcn_global_store_async_from_lds_b128)
#define ASYNC_LOAD128(g, l, off) \
  __builtin_amdgcn_global_load_async_to_lds_b128(to_glb(g), to_lds(l), (off), CPOL_RT)
#define ASYNC_STORE128(g, l, off) \
  __builtin_amdgcn_global_store_async_from_lds_b128(to_glb(g), to_lds(l), (off), CPOL_NT)
#else
#define ASYNC_LOAD128(g, l, off)                                            \
  asm volatile("global_load_async_to_lds_b128 %0, %1, off offset:%c2"       \
               :: "v"((uint32_t)(uintptr_t)(l)), "v"((const void*)(g)),     \
                  "i"(off) : "memory")
#define ASYNC_STORE128(g, l, off)                                           \
  asm volatile("global_store_async_from_lds_b128 %0, %1, off offset:%c2"    \
               :: "v"((const void*)(g)), "v"((uint32_t)(uintptr_t)(l)),     \
                  "i"(off) : "memory")
#endif

#if defined(__AMDGCN__) && __has_builtin(__builtin_amdgcn_s_wait_asynccnt)
#define WAIT_ASYNC0() __builtin_amdgcn_s_wait_asynccnt(0)
#else
#define WAIT_ASYNC0() asm volatile("s_wait_asynccnt 0" ::: "memory")
#endif

// ---------------- kernel ----------------
// Pure data movement: 1.31 GB streaming stores (HBM floor ~56 us @ 23.3 TB/s);
// node table (41 MB) stays L2-resident so gather reads never touch HBM.
// Each wave processes TWO (b,e) pairs per iteration through the CDNA5 async
// DMA path (no VGPR data movement):
//   8x global_load_async_to_lds_b128   (gather 2x{src,tgt} rows, 4 KB)
//   s_wait_asynccnt 0                  (also drains previous batch's stores,
//                                       making buffer-set reuse safe)
//   8x global_store_async_from_lds_b128 (NT-stream two 2 KB output rows)
// Two buffer sets ping-pong so loads(i) overlap stores(i-1).
__global__ __launch_bounds__(BLOCK_THREADS) void
NodePropagatorSparse_gather_async(const float* __restrict__ node,  // [B,N,D]
                                  const int*   __restrict__ esrc,  // [E]
                                  const int*   __restrict__ etgt,  // [E]
                                  float*       __restrict__ out)   // [B,E,2D]
{
  // 4 buffers of 2048 B per wave: {set0: buf0,buf1}, {set1: buf2,buf3}
  __shared__ alignas(16) unsigned char smem[WAVES_PER_BLOCK][4][2 * D_ * 4];

  const int lane  = threadIdx.x & (WAVE_SZ - 1);
  const int wid   = threadIdx.x >> 5;
  const int gwave = blockIdx.x * WAVES_PER_BLOCK + wid;
  const int nwave = GRID_BLOCKS * WAVES_PER_BLOCK;
  const int total = B_ * E_;

  int set = 0;
  for (int p0 = gwave; p0 < total; p0 += 2 * nwave, set ^= 1) {
    const int p1     = p0 + nwave;
    const bool has1  = (p1 < total);

    // ---- pair 0 addresses ----
    const int e0 = p0 % E_, b0 = p0 / E_;
    const int s0 = esrc[e0], t0 = etgt[e0];
    const char* src0 = (const char*)(node + ((size_t)b0 * N_ + s0) * D_) + lane * 16;
    const char* tgt0 = (const char*)(node + ((size_t)b0 * N_ + t0) * D_) + lane * 16 - 1024;
    unsigned char* l0 = &smem[wid][set * 2 + 0][0] + lane * 16;
    char* out0 = (char*)(out + (size_t)p0 * (2 * D_)) + lane * 16;

    ASYNC_LOAD128(src0, l0, 0);
    ASYNC_LOAD128(src0, l0, 512);
    ASYNC_LOAD128(tgt0, l0, 1024);
    ASYNC_LOAD128(tgt0, l0, 1536);

    // ---- pair 1 addresses (guarded) ----
    unsigned char* l1 = &smem[wid][set * 2 + 1][0] + lane * 16;
    char* out1 = nullptr;
    if (has1) {
      const int e1 = p1 % E_, b1 = p1 / E_;
      const int s1 = esrc[e1], t1 = etgt[e1];
      const char* src1 = (const char*)(node + ((size_t)b1 * N_ + s1) * D_) + lane * 16;
      const char* tgt1 = (const char*)(node + ((size_t)b1 * N_ + t1) * D_) + lane * 16 - 1024;
      out1 = (char*)(out + (size_t)p1 * (2 * D_)) + lane * 16;

      ASYNC_LOAD128(src1, l1, 0);
      ASYNC_LOAD128(src1, l1, 512);
      ASYNC_LOAD128(tgt1, l1, 1024);
      ASYNC_LOAD128(tgt1, l1, 1536);
    }

    // loads landed in LDS; previous batch's stores also drained
    WAIT_ASYNC0();

    ASYNC_STORE128(out0, l0, 0);
    ASYNC_STORE128(out0, l0, 512);
    ASYNC_STORE128(out0, l0, 1024);
    ASYNC_STORE128(out0, l0, 1536);
    if (has1) {
      ASYNC_STORE128(out1, l1, 0);
      ASYNC_STORE128(out1, l1, 512);
      ASYNC_STORE128(out1, l1, 1024);
      ASYNC_STORE128(out1, l1, 1536);
    }
  }
  // drain outstanding async stores before wave exit
  WAIT_ASYNC0();
}

// ---------------- launch ----------------
extern "C" void kernel_launch(void* const* d_in, const int* in_sizes, int n_in,
                              void* d_out, int out_size, void* d_ws, size_t ws_size,
                              hipStream_t stream) {
  const float* node = (const float*)d_in[0];   // [B,N,D] float32
  const int*   esrc = (const int*)d_in[1];     // [E] int
  const int*   etgt = (const int*)d_in[2];     // [E] int
  float*       out  = (float*)d_out;           // [B,E,2D] float32

  (void)in_sizes; (void)n_in; (void)out_size; (void)d_ws; (void)ws_size;

  NodePropagatorSparse_gather_async<<<GRID_BLOCKS, BLOCK_THREADS, 0, stream>>>(
      node, esrc, etgt, out);
}